// ConsumptionPredictor_47656957116472
// MI455X (gfx1250) — compile-verified
//
#include <hip/hip_runtime.h>
#include <hip/hip_bf16.h>
#include <cstddef>
#include <cstdint>

// ---------------------------------------------------------------------------
// Model: x(512,8,2048) -> conv1(16,k3)+relu -> conv2(12,k3)+relu ->
//        LSTM(H=5) -> LSTM(H=5) -> last step -> linear(5->1)
// Strategy:
//   K1: per (batch, 128-step tile): x slab pulled global->LDS with
//       GLOBAL_LOAD_ASYNC_TO_LDS_B32 (ASYNCcnt), conv1+conv2 in LDS (VALU),
//       then the layer-0 input projection as a GEMM on
//       V_WMMA_F32_16X16X4_F32 (K 12->16 padded, ones-row folds the bias;
//       N 20->32). Writes xg0 in (T, B, 20) layout, immediate-offset stores.
//   K2: fused 2-layer LSTM scan + final linear. One lane per batch row,
//       16 single-wave workgroups; recurrent weights in LDS (broadcast);
//       gate loads double-buffered + global_prefetch 8 steps ahead.
// ---------------------------------------------------------------------------

typedef __attribute__((ext_vector_type(2))) float v2f;
typedef __attribute__((ext_vector_type(8))) float v8f;

#define NB   512      // batch
#define NL   2048     // sequence length
#define LT   128      // timesteps per workgroup tile
#define NG   20       // 4*H gates

__device__ __forceinline__ float sigf(float x) {
  return 1.0f / (1.0f + __expf(-x));
}
__device__ __forceinline__ float tanhx(float x) {
  // (e^2x - 1)/(e^2x + 1); saturates correctly for |x| large
  float e = __expf(2.0f * x);
  return 1.0f - 2.0f / (e + 1.0f);
}

// ---------------------------------------------------------------------------
// Kernel 1: async x-slab load + conv1 + relu + conv2 + relu + WMMA projection
// grid = (NL/LT, NB), block = 128 (4 waves)
// ---------------------------------------------------------------------------
__global__ void __launch_bounds__(128)
conv_proj_kernel(const float* __restrict__ x,
                 const float* __restrict__ W1, const float* __restrict__ b1,
                 const float* __restrict__ W2, const float* __restrict__ b2,
                 const float* __restrict__ Wih0,
                 const float* __restrict__ bih0, const float* __restrict__ bhh0,
                 float* __restrict__ xg0)
{
  __shared__ float xs[8][132];        // x slab with halo (t0-2 .. t0+129)
  __shared__ float y1[16][130];       // conv1 out with halo (t0-1 .. t0+128)
  __shared__ float y2p[LT][16];       // conv2 out, K-padded: [12]=1 (bias), [13..15]=0
  __shared__ float Bp[16][32];        // Wih0^T padded: row12 = bih0+bhh0, N padded to 32
  __shared__ float sW1[16 * 24], sb1[16], sW2[12 * 48], sb2[12];

  const int tid = threadIdx.x;
  const int b   = blockIdx.y;
  const int t0  = blockIdx.x * LT;

  // x slab: per-lane async global->LDS copy (ASYNCcnt path); SAME-padding
  // slots are pre-zeroed with plain DS stores (disjoint addresses).
  for (int i = tid; i < 8 * 132; i += 128) {
    int c = i / 132, p = i % 132;
    int gt = t0 - 2 + p;
    if (gt >= 0 && gt < NL) {
      unsigned lds_addr = (unsigned)(uintptr_t)&xs[c][p];
      const float* gptr = &x[((size_t)b * 8 + c) * NL + gt];
      asm volatile("global_load_async_to_lds_b32 %0, %1, off"
                   :: "v"(lds_addr), "v"(gptr) : "memory");
    } else {
      xs[c][p] = 0.0f;
    }
  }

  for (int i = tid; i < 384; i += 128) sW1[i] = W1[i];
  for (int i = tid; i < 576; i += 128) sW2[i] = W2[i];
  if (tid < 16) sb1[tid] = b1[tid];
  if (tid < 12) sb2[tid] = b2[tid];

  // B matrix (16x32): Bp[k][g] = Wih0[g][k] (k<12), bias row at k==12, else 0
  for (int i = tid; i < 16 * 32; i += 128) {
    int k = i >> 5, g = i & 31;
    float v = 0.0f;
    if (g < NG) {
      if (k < 12)       v = Wih0[g * 12 + k];
      else if (k == 12) v = bih0[g] + bhh0[g];
    }
    Bp[k][g] = v;
  }

  asm volatile("s_wait_asynccnt 0x0" ::: "memory");   // x slab resident in LDS
  __syncthreads();

  // conv1 + relu (16 ch x 130 positions incl. halo); zero outside [0, NL)
  for (int i = tid; i < 16 * 130; i += 128) {
    int o = i / 130, tt = i % 130;
    int gt = t0 - 1 + tt;
    float s = sb1[o];
#pragma unroll
    for (int c = 0; c < 8; ++c)
#pragma unroll
      for (int k = 0; k < 3; ++k)
        s += xs[c][tt + k] * sW1[(o * 8 + c) * 3 + k];
    y1[o][tt] = (gt >= 0 && gt < NL) ? fmaxf(s, 0.0f) : 0.0f;
  }
  __syncthreads();

  // conv2 + relu, stored t-major and K-padded for the WMMA A operand
  for (int i = tid; i < LT * 16; i += 128) {
    int tt = i >> 4, j = i & 15;
    float v;
    if (j < 12) {
      float s = sb2[j];
#pragma unroll
      for (int c = 0; c < 16; ++c)
#pragma unroll
        for (int k = 0; k < 3; ++k)
          s += y1[c][tt + k] * sW2[(j * 16 + c) * 3 + k];
      v = fmaxf(s, 0.0f);
    } else {
      v = (j == 12) ? 1.0f : 0.0f;    // ones column folds the bias row of Bp
    }
    y2p[tt][j] = v;
  }
  __syncthreads();

  // WMMA projection: D(16t x 16g) += A(16t x 4k) * B(4k x 16g), 4 K-chunks.
  // ISA layouts: A lane holds M=lane&15, K=2*(lane>>4)+vgpr; B mirrored on N;
  // C vgpr r: M=r+8*(lane>>4), N=lane&15.
  const int lane = tid & 31, wave = tid >> 5;
  const int hi = lane >> 4, lm = lane & 15;

  for (int mt = wave * 2; mt <= wave * 2 + 1; ++mt) {
    const int trow = mt * 16 + lm;
    for (int nt = 0; nt < 2; ++nt) {
      v8f acc = {};
#pragma unroll
      for (int c = 0; c < 4; ++c) {
        const int kb = c * 4 + 2 * hi;
        v2f a, bb;
        a.x  = y2p[trow][kb];
        a.y  = y2p[trow][kb + 1];
        bb.x = Bp[kb][nt * 16 + lm];
        bb.y = Bp[kb + 1][nt * 16 + lm];
        acc = __builtin_amdgcn_wmma_f32_16x16x4_f32(
            /*neg_a=*/false, a, /*neg_b=*/false, bb,
            /*c_mod=*/(short)0, acc, /*reuse_a=*/false, /*reuse_b=*/false);
      }
      const int g = nt * 16 + lm;
      // one base address, 8 stores at constant offsets r*NB*NG*4 (= r*40960 B,
      // fits the 24-bit signed instruction offset -> immediate-offset stores)
      float* pout = xg0 + ((size_t)(t0 + mt * 16 + 8 * hi) * NB + b) * NG + g;
      if (g < NG) {
#pragma unroll
        for (int r = 0; r < 8; ++r) pout[r * (NB * NG)] = acc[r];
      }
    }
  }
}

// ---------------------------------------------------------------------------
// Kernel 2: fused 2-layer LSTM scan + final linear. One lane per batch row.
// grid = 16 blocks x 32 threads (single wave per WG -> barriers are NOPs,
// 16 waves spread over 16 SIMD32s). Weights in LDS (same-address broadcast).
// Gate stream double-buffered + prefetched 8 steps ahead.
// ---------------------------------------------------------------------------
__global__ void __launch_bounds__(32)
lstm_scan_kernel(const float* __restrict__ xg0,
                 const float* __restrict__ Whh0,
                 const float* __restrict__ Wih1, const float* __restrict__ Whh1,
                 const float* __restrict__ bih1, const float* __restrict__ bhh1,
                 const float* __restrict__ Wlin, const float* __restrict__ blin,
                 float* __restrict__ out)
{
  __shared__ float sA[100];   // Whh0[g][k]
  __shared__ float sB[100];   // Wih1[g][k]
  __shared__ float sC[100];   // Whh1[g][k]
  __shared__ float sb[20];    // bih1 + bhh1
  __shared__ float sl[6];     // Wlin[0..4], blin

  const int lane = threadIdx.x;
  for (int i = lane; i < 100; i += 32) {
    sA[i] = Whh0[i];
    sB[i] = Wih1[i];
    sC[i] = Whh1[i];
  }
  if (lane < 20) sb[lane] = bih1[lane] + bhh1[lane];
  if (lane < 5)  sl[lane] = Wlin[lane];
  if (lane == 5) sl[5] = blin[0];
  __syncthreads();

  const int b = blockIdx.x * 32 + lane;
  const float* bbase = xg0 + (size_t)b * NG;   // (T, B, 20): step stride NB*NG

  float h0[5] = {}, c0[5] = {}, h1[5] = {}, c1[5] = {};

  // prologue load for t = 0
  float4 u0, u1, u2, u3, u4;
  {
    const float4* p = (const float4*)bbase;
    u0 = p[0]; u1 = p[1]; u2 = p[2]; u3 = p[3]; u4 = p[4];
  }

  for (int t = 0; t < NL; ++t) {
    // issue next step's loads before this step's ~600 VALU ops (latency hide)
    const int tn = (t + 1 < NL) ? t + 1 : NL - 1;
    const float4* pn = (const float4*)(bbase + (size_t)tn * (NB * NG));
    float4 n0 = pn[0], n1 = pn[1], n2 = pn[2], n3 = pn[3], n4 = pn[4];
    const int tp = (t + 8 < NL) ? t + 8 : NL - 1;
    __builtin_prefetch(bbase + (size_t)tp * (NB * NG), 0, 1);

    float g[20] = { u0.x, u0.y, u0.z, u0.w, u1.x, u1.y, u1.z, u1.w,
                    u2.x, u2.y, u2.z, u2.w, u3.x, u3.y, u3.z, u3.w,
                    u4.x, u4.y, u4.z, u4.w };

#pragma unroll
    for (int j = 0; j < 20; ++j) {
      float s = g[j];
#pragma unroll
      for (int k = 0; k < 5; ++k) s += h0[k] * sA[j * 5 + k];
      g[j] = s;
    }
#pragma unroll
    for (int k = 0; k < 5; ++k) {
      float ii = sigf(g[k]);
      float ff = sigf(g[5 + k]);
      float gg = tanhx(g[10 + k]);
      float oo = sigf(g[15 + k]);
      c0[k] = ff * c0[k] + ii * gg;
      h0[k] = oo * tanhx(c0[k]);
    }

    float q[20];
#pragma unroll
    for (int j = 0; j < 20; ++j) {
      float s = sb[j];
#pragma unroll
      for (int k = 0; k < 5; ++k) s += sB[j * 5 + k] * h0[k] + sC[j * 5 + k] * h1[k];
      q[j] = s;
    }
#pragma unroll
    for (int k = 0; k < 5; ++k) {
      float ii = sigf(q[k]);
      float ff = sigf(q[5 + k]);
      float gg = tanhx(q[10 + k]);
      float oo = sigf(q[15 + k]);
      c1[k] = ff * c1[k] + ii * gg;
      h1[k] = oo * tanhx(c1[k]);
    }

    u0 = n0; u1 = n1; u2 = n2; u3 = n3; u4 = n4;
  }

  float s = sl[5];
#pragma unroll
  for (int k = 0; k < 5; ++k) s += h1[k] * sl[k];
  out[b] = s;
}

// ---------------------------------------------------------------------------
extern "C" void kernel_launch(void* const* d_in, const int* in_sizes, int n_in,
                              void* d_out, int out_size, void* d_ws, size_t ws_size,
                              hipStream_t stream) {
  const float* x    = (const float*)d_in[0];
  const float* W1   = (const float*)d_in[1];
  const float* b1   = (const float*)d_in[2];
  const float* W2   = (const float*)d_in[3];
  const float* b2   = (const float*)d_in[4];
  const float* Wih0 = (const float*)d_in[5];
  const float* Whh0 = (const float*)d_in[6];
  const float* bih0 = (const float*)d_in[7];
  const float* bhh0 = (const float*)d_in[8];
  const float* Wih1 = (const float*)d_in[9];
  const float* Whh1 = (const float*)d_in[10];
  const float* bih1 = (const float*)d_in[11];
  const float* bhh1 = (const float*)d_in[12];
  const float* Wlin = (const float*)d_in[13];
  const float* blin = (const float*)d_in[14];

  float* xg0 = (float*)d_ws;    // (T, B, 20) f32 = 80 MB

  dim3 g1(NL / LT, NB);
  conv_proj_kernel<<<g1, 128, 0, stream>>>(x, W1, b1, W2, b2, Wih0, bih0, bhh0, xg0);
  lstm_scan_kernel<<<16, 32, 0, stream>>>(xg0, Whh0, Wih1, Whh1, bih1, bhh1,
                                          Wlin, blin, (float*)d_out);
}